// GaussianQuantumNeuralNetwork_18296560681019
// MI455X (gfx1250) — compile-verified
//
#include <hip/hip_runtime.h>
#include <math.h>

// ---------------------------------------------------------------------------
// GaussianQuantumNeuralNetwork forward for MI455X (gfx1250, wave32, WMMA).
//
// Key algebraic collapse: the whole circuit is linear in psi, so we build a
// single 100x100 complex operator Mtot on device (matrix exponentials via
// scaling-and-squaring Taylor), and the batch pass becomes two real GEMMs
// (psi_init is real):  psi_r = X @ Mr^T,  psi_i = X @ Mi^T, executed with
// V_WMMA_F32_16X16X4_F32 (f32 to match the complex64 reference).
// ---------------------------------------------------------------------------

#define CUT 10
#define DBIG 100
#define DPAD 112
#define MAXL 4

typedef float v2f __attribute__((ext_vector_type(2)));
typedef float v8f __attribute__((ext_vector_type(8)));

// ---- device-global scratch (recomputed deterministically every launch) ----
__device__ float2 g_U1[MAXL][DBIG*DBIG];
__device__ float2 g_U2[MAXL][DBIG*DBIG];
__device__ float2 g_S1[MAXL][CUT*CUT];
__device__ float2 g_S2[MAXL][CUT*CUT];
__device__ float2 g_D1[MAXL][CUT*CUT];
__device__ float2 g_D2[MAXL][CUT*CUT];
__device__ float2 g_pool[6*MAXL][30000];     // per-block expm scratch (G,P,T)
__device__ float2 g_Mlayer[MAXL][DBIG*DBIG];
__device__ float2 g_Mtot[DBIG*DBIG];
__device__ float2 g_Mtmp[DBIG*DBIG];
__device__ float  g_MtR[DPAD*DPAD];          // B-operand: MtR[k][j] = Re Mtot[j][k]
__device__ float  g_MtI[DPAD*DPAD];

__constant__ float INV_SQRT_FACT[CUT] = {
  1.0f, 1.0f, 0.70710678f, 0.40824829f, 0.20412415f,
  0.09128709f, 0.03726780f, 0.01408591f, 0.00498012f, 0.00166004f };
__constant__ float SQN[CUT-1] = {
  1.0f, 1.41421356f, 1.73205081f, 2.0f, 2.23606798f,
  2.44948975f, 2.64575131f, 2.82842712f, 3.0f };

__device__ __forceinline__ float2 cmul(float2 a, float2 b) {
  return make_float2(a.x*b.x - a.y*b.y, a.x*b.y + a.y*b.x);
}
__device__ __forceinline__ float2 cis(float t) {
  return make_float2(cosf(t), sinf(t));
}

// Block-cooperative complex GEMM: C = scale*(A*B) (+I).  Row-major, dim d.
__device__ void cgemm_block(const float2* A, const float2* Bm, float2* C,
                            int d, float scale, bool addI) {
  for (int idx = threadIdx.x; idx < d*d; idx += blockDim.x) {
    const int i = idx / d, j = idx % d;
    float ar = 0.f, ai = 0.f;
    for (int k = 0; k < d; ++k) {
      float2 a = A[i*d + k], b = Bm[k*d + j];
      ar += a.x*b.x - a.y*b.y;
      ai += a.x*b.y + a.y*b.x;
    }
    float2 c; c.x = ar*scale; c.y = ai*scale;
    if (addI && i == j) c.x += 1.f;
    C[idx] = c;
  }
}

// Block-cooperative expm(G) -> dst.  Scaling & squaring + 12-term Horner.
__device__ void block_expm(float2* G, float2* P, float2* T, float2* dst, int d) {
  __shared__ float red[256];
  float mx = 0.f;
  for (int i = threadIdx.x; i < d; i += blockDim.x) {
    float s = 0.f;
    for (int j = 0; j < d; ++j) { float2 g = G[i*d + j]; s += fabsf(g.x) + fabsf(g.y); }
    mx = fmaxf(mx, s);
  }
  red[threadIdx.x] = mx;
  __syncthreads();
  if (threadIdx.x == 0) {
    float m2 = 0.f;
    for (unsigned t = 0; t < blockDim.x; ++t) m2 = fmaxf(m2, red[t]);
    red[0] = m2;
  }
  __syncthreads();
  float nrm = red[0];
  int s = 0;
  while (nrm > 0.5f && s < 30) { nrm *= 0.5f; ++s; }
  const float sc = ldexpf(1.f, -s);
  __syncthreads();
  for (int idx = threadIdx.x; idx < d*d; idx += blockDim.x) {
    G[idx].x *= sc; G[idx].y *= sc;
  }
  __syncthreads();
  const int NT = 12;
  for (int idx = threadIdx.x; idx < d*d; idx += blockDim.x) {
    float2 v = G[idx]; v.x *= (1.f/NT); v.y *= (1.f/NT);
    if (idx / d == idx % d) v.x += 1.f;
    P[idx] = v;
  }
  __syncthreads();
  float2 *p = P, *t = T;
  for (int k = NT - 1; k >= 1; --k) {      // p = I + G*p/k
    cgemm_block(G, p, t, d, 1.f/(float)k, true);
    __syncthreads();
    float2* tmp = p; p = t; t = tmp;
  }
  for (int q = 0; q < s; ++q) {            // repeated squaring
    cgemm_block(p, p, t, d, 1.f, false);
    __syncthreads();
    float2* tmp = p; p = t; t = tmp;
  }
  for (int idx = threadIdx.x; idx < d*d; idx += blockDim.x) dst[idx] = p[idx];
}

// ---------------------------------------------------------------------------
// Kernel 1: each block builds one generator and exponentiates it.
// block b: kind = b / NL (0:U1 1:U2 2:S1 3:S2 4:D1 5:D2), layer = b % NL.
// ---------------------------------------------------------------------------
__global__ __launch_bounds__(128)
void gqnn_expm_kernel(const float* __restrict__ qw, int NL) {
  const int b = blockIdx.x;
  const int kind = b / NL, layer = b % NL;
  const float* w = qw + layer * 16;
  float2* G = &g_pool[b][0];
  float2* P = &g_pool[b][10000];
  float2* T = &g_pool[b][20000];
  int d;
  float2* dst;

  if (kind <= 1) {                                 // beamsplitter, 100x100
    d = DBIG;
    const float th = w[kind == 0 ? 0 : 8];
    const float ph = w[kind == 0 ? 1 : 9];
    const float cp = cosf(ph), sp = sinf(ph);
    for (int idx = threadIdx.x; idx < DBIG*DBIG; idx += blockDim.x) {
      const int R = idx / DBIG, C = idx % DBIG;
      const int n = R/10, m = R%10, k = C/10, l = C%10;
      // AD[n,k]*A[m,l]  and  A[n,k]*AD[m,l]
      const float t1 = ((n == k+1) ? sqrtf((float)n) : 0.f) *
                       ((l == m+1) ? sqrtf((float)l) : 0.f);
      const float t2 = ((k == n+1) ? sqrtf((float)k) : 0.f) *
                       ((m == l+1) ? sqrtf((float)m) : 0.f);
      G[idx] = make_float2(th*cp*(t1 - t2), th*sp*(t1 + t2));
    }
    dst = (kind == 0) ? g_U1[layer] : g_U2[layer];
  } else if (kind <= 3) {                          // squeezer, 10x10
    d = CUT;
    const float r  = w[kind == 2 ? 4 : 6];
    const float ph = w[kind == 2 ? 5 : 7];
    const float zr = r*cosf(ph), zi = r*sinf(ph);
    for (int idx = threadIdx.x; idx < CUT*CUT; idx += blockDim.x) {
      const int i = idx / CUT, j = idx % CUT;
      const float a2  = (j == i+2) ? sqrtf((float)((i+1)*(i+2))) : 0.f;  // (A@A)
      const float ad2 = (i == j+2) ? sqrtf((float)((j+1)*(j+2))) : 0.f;  // (AD@AD)
      G[idx] = make_float2(0.5f*zr*(a2 - ad2), -0.5f*zi*(a2 + ad2));
    }
    dst = (kind == 2) ? g_S1[layer] : g_S2[layer];
  } else {                                         // displacement, 10x10
    d = CUT;
    const float r = w[kind == 4 ? 12 : 13];
    for (int idx = threadIdx.x; idx < CUT*CUT; idx += blockDim.x) {
      const int i = idx / CUT, j = idx % CUT;
      const float ad = (i == j+1) ? sqrtf((float)i) : 0.f;
      const float a  = (j == i+1) ? sqrtf((float)j) : 0.f;
      G[idx] = make_float2(r*(ad - a), 0.f);
    }
    dst = (kind == 4) ? g_D1[layer] : g_D2[layer];
  }
  __syncthreads();
  block_expm(G, P, T, dst, d);
}

// ---------------------------------------------------------------------------
// Kernel 2: per layer, fold phases + kron products + chain:
//   Mlayer = diag(Kerr)*kron(D1,D2)*diag(P10,11)*U2 * kron(S1,S2)*diag(P2,3)*U1
// ---------------------------------------------------------------------------
__global__ __launch_bounds__(256)
void gqnn_compose_layer_kernel(const float* __restrict__ qw) {
  const int l = blockIdx.x;
  const float* w = qw + l * 16;
  float2* K  = &g_pool[2*l][0];
  float2* W  = &g_pool[2*l][10000];
  float2* M1 = &g_pool[2*l + 1][0];
  float2* M2 = &g_pool[2*l + 1][10000];

  for (int idx = threadIdx.x; idx < DBIG*DBIG; idx += blockDim.x) {
    const int R = idx / DBIG, C = idx % DBIG;
    const int n = R/10, m = R%10, k = C/10, l2 = C%10;
    W[idx] = cmul(cis(w[2]*(float)n + w[3]*(float)m), g_U1[l][idx]);   // diag(P23)*U1
    K[idx] = cmul(g_S1[l][n*CUT + k], g_S2[l][m*CUT + l2]);            // kron(S1,S2)
  }
  __syncthreads();
  cgemm_block(K, W, M1, DBIG, 1.f, false);
  __syncthreads();
  for (int idx = threadIdx.x; idx < DBIG*DBIG; idx += blockDim.x) {
    const int R = idx / DBIG, C = idx % DBIG;
    const int n = R/10, m = R%10, k = C/10, l2 = C%10;
    W[idx] = cmul(cis(w[10]*(float)n + w[11]*(float)m), g_U2[l][idx]); // diag(P1011)*U2
    K[idx] = cmul(g_D1[l][n*CUT + k], g_D2[l][m*CUT + l2]);            // kron(D1,D2)
  }
  __syncthreads();
  cgemm_block(K, W, M2, DBIG, 1.f, false);
  __syncthreads();
  for (int idx = threadIdx.x; idx < DBIG*DBIG; idx += blockDim.x) {    // Kerr rows
    const int R = idx / DBIG;
    const int n = R/10, m = R%10;
    M2[idx] = cmul(cis(w[14]*(float)(n*n) + w[15]*(float)(m*m)), M2[idx]);
  }
  __syncthreads();
  cgemm_block(M2, M1, g_Mlayer[l], DBIG, 1.f, false);
}

// ---------------------------------------------------------------------------
// Kernel 3: Mtot = Mlayer[NL-1] * ... * Mlayer[0]; write padded transposed
// real/imag planes for the WMMA B-operand: Mt?[k*112+j] = Mtot[j][k].
// ---------------------------------------------------------------------------
__global__ __launch_bounds__(256)
void gqnn_compose_total_kernel(int NL) {
  float2* cur = g_Mtot;
  float2* tmp = g_Mtmp;
  for (int idx = threadIdx.x; idx < DBIG*DBIG; idx += blockDim.x)
    cur[idx] = g_Mlayer[0][idx];
  __syncthreads();
  for (int l = 1; l < NL; ++l) {
    cgemm_block(g_Mlayer[l], cur, tmp, DBIG, 1.f, false);
    __syncthreads();
    float2* t = cur; cur = tmp; tmp = t;
  }
  for (int idx = threadIdx.x; idx < DPAD*DPAD; idx += blockDim.x) {
    const int k = idx / DPAD, j = idx % DPAD;
    float2 v = (k < DBIG && j < DBIG) ? cur[j*DBIG + k] : make_float2(0.f, 0.f);
    g_MtR[idx] = v.x;
    g_MtI[idx] = v.y;
  }
}

// ---------------------------------------------------------------------------
// Kernel 4: batch pass. 128 threads = 4 waves; each wave owns 16 batch rows.
// psi_r/psi_i via V_WMMA_F32_16X16X4_F32 (K padded to 112, N = 7 tiles).
// ISA wave32 fragment layouts (CDNA5 ISA):
//   A 16x4 : lane m=L&15; VGPR0=K(2*half), VGPR1=K(2*half+1)
//   B 4x16 : lane n=L&15; VGPR0=K(2*half) row, VGPR1=K(2*half+1) row
//   C 16x16: VGPR r -> row r+8*half, col L&15
// ---------------------------------------------------------------------------
__global__ __launch_bounds__(128)
void gqnn_batch_kernel(const float* __restrict__ x, float* __restrict__ out, int B) {
  __shared__ float sA[4][16][DPAD];   // xvec, later psi_r   (28672 B)
  __shared__ float sB[4][16][DPAD];   // coherent c's, later psi_i
  __shared__ float sM[2][4][DPAD];    // staged operator k-rows (re,im)

  const int tid = threadIdx.x;
  const int w   = tid >> 5;
  const int L   = tid & 31;
  const int lane15 = L & 15;
  const int hlf = L >> 4;
  const int rb  = blockIdx.x * 64 + w * 16;

  // coherent-state amplitudes: lanes 0-15 -> mode 0, lanes 16-31 -> mode 1
  {
    const int s = lane15, which = hlf, b = rb + s;
    const bool ok = (b < B);
    const float alpha = ok ? x[b*2 + which] : 0.f;
    const float e = expf(-0.5f*alpha*alpha);
    float p = 1.f;
    for (int n = 0; n < CUT; ++n) {
      sB[w][s][which*CUT + n] = ok ? (e * p * INV_SQRT_FACT[n]) : 0.f;
      p *= alpha;
    }
  }
  __syncthreads();
  // xvec[j] = c0[j/10]*c1[j%10], zero-padded 100..111  (real initial state)
  for (int idx = L; idx < 16*DPAD; idx += 32) {
    const int s = idx / DPAD, j = idx % DPAD;
    sA[w][s][j] = (j < DBIG) ? sB[w][s][j/10] * sB[w][s][CUT + j%10] : 0.f;
  }
  __syncthreads();

  const v8f vzero = {0.f,0.f,0.f,0.f,0.f,0.f,0.f,0.f};
  v8f accR[7], accI[7];
  for (int nt = 0; nt < 7; ++nt) { accR[nt] = vzero; accI[nt] = vzero; }

  for (int kt = 0; kt < 28; ++kt) {
    const int k0 = kt * 4;
    // stage 4 operator rows (re+im) into LDS, shared by all 4 waves
    for (int idx = tid; idx < 2*4*DPAD; idx += 128) {
      const int pl = idx / (4*DPAD), r2 = idx % (4*DPAD);
      const int kr = r2 / DPAD, cj = r2 % DPAD;
      const float* src = pl ? g_MtI : g_MtR;
      sM[pl][kr][cj] = src[(k0 + kr)*DPAD + cj];
    }
    __syncthreads();
    v2f a;
    a.x = sA[w][lane15][k0 + 2*hlf];
    a.y = sA[w][lane15][k0 + 2*hlf + 1];
    for (int nt = 0; nt < 7; ++nt) {
      const int col = nt*16 + lane15;
      v2f br, bi;
      br.x = sM[0][2*hlf][col];  br.y = sM[0][2*hlf + 1][col];
      bi.x = sM[1][2*hlf][col];  bi.y = sM[1][2*hlf + 1][col];
      accR[nt] = __builtin_amdgcn_wmma_f32_16x16x4_f32(
          false, a, false, br, (short)0, accR[nt], false, false);
      accI[nt] = __builtin_amdgcn_wmma_f32_16x16x4_f32(
          false, a, false, bi, (short)0, accI[nt], false, false);
    }
    __syncthreads();
  }

  // write psi back to LDS (overwriting xvec / coherent buffers)
  for (int nt = 0; nt < 7; ++nt) {
    for (int r = 0; r < 8; ++r) {
      const int row = r + 8*hlf, col = nt*16 + lane15;
      sA[w][row][col] = accR[nt][r];
      sB[w][row][col] = accI[nt][r];
    }
  }
  __syncthreads();

  // <X> expectations from tridiagonal X: e = 2*sum sqrt(n+1)*Re(psi*[n] psi[n+1])
  {
    const int s = L >> 1, which = L & 1;
    float acc = 0.f;
    if (which == 0) {                 // mode 0: pair rows n, n+1 (j stride 10)
      for (int n = 0; n < CUT-1; ++n) {
        const float wn = SQN[n];
        for (int m2 = 0; m2 < CUT; ++m2) {
          const int j0 = n*CUT + m2, j1 = j0 + CUT;
          acc += wn * (sA[w][s][j0]*sA[w][s][j1] + sB[w][s][j0]*sB[w][s][j1]);
        }
      }
    } else {                          // mode 1: pair cols m, m+1 (j stride 1)
      for (int n = 0; n < CUT; ++n) {
        for (int m2 = 0; m2 < CUT-1; ++m2) {
          const int j0 = n*CUT + m2, j1 = j0 + 1;
          acc += SQN[m2] * (sA[w][s][j0]*sA[w][s][j1] + sB[w][s][j0]*sB[w][s][j1]);
        }
      }
    }
    const int b = rb + s;
    if (b < B) out[b*2 + which] = 2.f * acc;
  }
}

// ---------------------------------------------------------------------------
extern "C" void kernel_launch(void* const* d_in, const int* in_sizes, int n_in,
                              void* d_out, int out_size, void* d_ws, size_t ws_size,
                              hipStream_t stream) {
  (void)n_in; (void)d_ws; (void)ws_size; (void)out_size;
  const float* x  = (const float*)d_in[0];
  const float* qw = (const float*)d_in[1];
  const int B  = in_sizes[0] / 2;
  int NL = in_sizes[1] / 16;
  if (NL < 1) NL = 1;
  if (NL > MAXL) NL = MAXL;

  gqnn_expm_kernel<<<6*NL, 128, 0, stream>>>(qw, NL);
  gqnn_compose_layer_kernel<<<NL, 256, 0, stream>>>(qw);
  gqnn_compose_total_kernel<<<1, 256, 0, stream>>>(NL);
  gqnn_batch_kernel<<<(B + 63)/64, 128, 0, stream>>>(x, (float*)d_out, B);
}